// DGCNN_16329465660218
// MI455X (gfx1250) — compile-verified
//
#include <hip/hip_runtime.h>

#define BATCH 8
#define NPTS  2048
#define KNN   20
#define NBITS 11            // log2(NPTS)
#define EPSV  1e-5f

typedef float v2f __attribute__((ext_vector_type(2)));
typedef float v8f __attribute__((ext_vector_type(8)));

// ---------------------------------------------------------------------------
// Transpose the (B,3,N) network input to (B,N,3) once.
// ---------------------------------------------------------------------------
__global__ void transpose3_kernel(const float* __restrict__ x, float* __restrict__ xt) {
    int t = blockIdx.x * blockDim.x + threadIdx.x;
    if (t >= BATCH * NPTS) return;
    int b = t >> NBITS, n = t & (NPTS - 1);
    const float* p = x + (size_t)b * 3 * NPTS + n;
    xt[(size_t)t * 3 + 0] = p[0];
    xt[(size_t)t * 3 + 1] = p[NPTS];
    xt[(size_t)t * 3 + 2] = p[2 * NPTS];
}

// ---------------------------------------------------------------------------
// Squared norms per point from transposed layout (contiguous row read).
// ---------------------------------------------------------------------------
__global__ void xx_kernel(const float* __restrict__ xt, float* __restrict__ xx, int C) {
    int t = blockIdx.x * blockDim.x + threadIdx.x;
    if (t >= BATCH * NPTS) return;
    const float* p = xt + (size_t)t * C;
    float s = 0.f;
    for (int c = 0; c < C; ++c) s = fmaf(p[c], p[c], s);
    xx[t] = s;
}

// ---------------------------------------------------------------------------
// kNN: one wave per query point. Distances in LDS, 20x wave-argmax with
// low-index tie break (matches jax.lax.top_k). xt is (B,N,C) so the dot
// product streams contiguous float4s.
// ---------------------------------------------------------------------------
template<int C>
__global__ __launch_bounds__(128)
void knn_kernel(const float* __restrict__ xt, const float* __restrict__ xx,
                int* __restrict__ idx) {
    __shared__ float dist[4][NPTS];
    __shared__ __align__(16) float xq[4][128];
    const int wv = threadIdx.x >> 5, l = threadIdx.x & 31;
    const int row = blockIdx.x * 4 + wv;
    const int b = row >> NBITS;

    const float* rowq = xt + (size_t)row * C;
    for (int c = l; c < C; c += 32) xq[wv][c] = rowq[c];
    __syncthreads();

    const float xxq = xx[row];
    for (int m = l; m < NPTS; m += 32) {
        const float* pm = xt + (size_t)(b * NPTS + m) * C;
        float dot = 0.f;
        if (C % 4 == 0) {
            #pragma unroll
            for (int c = 0; c < C; c += 4) {
                float4 v = *(const float4*)(pm + c);
                dot = fmaf(xq[wv][c + 0], v.x, dot);
                dot = fmaf(xq[wv][c + 1], v.y, dot);
                dot = fmaf(xq[wv][c + 2], v.z, dot);
                dot = fmaf(xq[wv][c + 3], v.w, dot);
            }
        } else {
            #pragma unroll
            for (int c = 0; c < C; ++c) dot = fmaf(xq[wv][c], pm[c], dot);
        }
        dist[wv][m] = 2.f * dot - xxq - xx[b * NPTS + m];
    }
    __syncthreads();

    for (int t = 0; t < KNN; ++t) {
        float bv = -__builtin_inff(); int bi = NPTS;
        for (int m = l; m < NPTS; m += 32) {
            float v = dist[wv][m];
            if (v > bv || (v == bv && m < bi)) { bv = v; bi = m; }
        }
        for (int off = 16; off > 0; off >>= 1) {
            float ov = __shfl_xor(bv, off, 32);
            int   oi = __shfl_xor(bi, off, 32);
            if (ov > bv || (ov == bv && oi < bi)) { bv = ov; bi = oi; }
        }
        if (l == 0) { idx[(size_t)row * KNN + t] = bi; dist[wv][bi] = -__builtin_inff(); }
        __syncthreads();
    }
}

// ---------------------------------------------------------------------------
// Fused EdgeConv GEMM via V_WMMA_F32_16X16X4_F32, transposed feature layout.
//   y = W_nbr*(x_nbr - x_ctr) + W_ctr*x_ctr
// The second term is K-invariant -> computed once into acc0; A-fragments and
// center features of the neighbor half are hoisted to registers, so the
// steady-state K-loop body is: 1x b64 load + v_pk sub + 1x wmma per step.
// PASS 0: per-channel sum/sumsq (BN stats). PASS 1: lrelu(scale*y+shift),
// max over K, contiguous store into (B,N,COUT).
// ---------------------------------------------------------------------------
template<int CH, int COUT, int PASS>
__global__ __launch_bounds__(128)
void edge_conv_kernel(const float* __restrict__ xt, const float* __restrict__ w,
                      const int* __restrict__ idx, float* __restrict__ stats,
                      const float* __restrict__ ss, float* __restrict__ outT) {
    constexpr int CIN    = 2 * CH;
    constexpr int KSTEPS = (CIN + 3) / 4;
    constexpr int S1     = (CH % 4 == 0) ? CH / 4 : 0;   // neighbor-half steps
    constexpr int S1A    = (S1 > 0) ? S1 : 1;
    constexpr int CT     = (BATCH * NPTS) / 16;          // 1024 column tiles

    const int l  = threadIdx.x & 31;
    const int W  = blockIdx.x * 4 + (threadIdx.x >> 5);
    const int ro = (W / CT) * 16;
    const int nIdx = (W % CT) * 16 + (l & 15);
    const int b = nIdx >> NBITS;
    const int kh = l >> 4;

    const float* wrow = w + (size_t)(ro + (l & 15)) * CIN;
    const float* ptc  = xt + (size_t)nIdx * CH;           // center row
    const int*   ip   = idx + (size_t)nIdx * KNN;

    float s1[8], s2[8], mx[8], sc[8], sh[8];
    #pragma unroll
    for (int i = 0; i < 8; ++i) { s1[i] = 0.f; s2[i] = 0.f; mx[i] = -__builtin_inff(); }
    if (PASS == 1) {
        #pragma unroll
        for (int i = 0; i < 8; ++i) { int r = ro + 8 * kh + i; sc[i] = ss[2 * r]; sh[i] = ss[2 * r + 1]; }
    }

    if (CH % 4 == 0) {
        // ---- K-invariant half: channels [CH, 2CH) use center features ----
        v8f acc0 = {};
        #pragma unroll
        for (int s = S1; s < KSTEPS; ++s) {
            const int c0 = 4 * s + 2 * kh;
            v2f a  = *(const v2f*)(wrow + c0);
            v2f bm = *(const v2f*)(ptc + (c0 - CH));
            acc0 = __builtin_amdgcn_wmma_f32_16x16x4_f32(false, a, false, bm,
                                                         (short)0, acc0, false, false);
        }
        // ---- hoist A-fragments + center features for neighbor half ----
        v2f aR[S1A], cR[S1A];
        #pragma unroll
        for (int s = 0; s < S1; ++s) {
            const int c0 = 4 * s + 2 * kh;
            aR[s] = *(const v2f*)(wrow + c0);
            cR[s] = *(const v2f*)(ptc + c0);
        }
        for (int kk = 0; kk < KNN; ++kk) {
            const int nbr = ip[kk];
            const float* ptn = xt + (size_t)(b * NPTS + nbr) * CH;
            v8f acc = acc0;
            #pragma unroll
            for (int s = 0; s < S1; ++s) {
                const int c0 = 4 * s + 2 * kh;
                v2f v = *(const v2f*)(ptn + c0);
                v2f bm; bm.x = v.x - cR[s].x; bm.y = v.y - cR[s].y;
                acc = __builtin_amdgcn_wmma_f32_16x16x4_f32(false, aR[s], false, bm,
                                                            (short)0, acc, false, false);
            }
            if (PASS == 0) {
                #pragma unroll
                for (int i = 0; i < 8; ++i) { float v = acc[i]; s1[i] += v; s2[i] += v * v; }
            } else {
                #pragma unroll
                for (int i = 0; i < 8; ++i) {
                    float v = acc[i] * sc[i] + sh[i];
                    v = (v >= 0.f) ? v : 0.2f * v;
                    mx[i] = fmaxf(mx[i], v);
                }
            }
        }
    } else {
        // ---- generic (block 1, CH=3): branchless clamped-index gather ----
        for (int kk = 0; kk < KNN; ++kk) {
            const int nbr = ip[kk];
            const float* ptn = xt + (size_t)(b * NPTS + nbr) * CH;
            v8f acc = {};
            #pragma unroll
            for (int s = 0; s < KSTEPS; ++s) {
                const int c0 = 4 * s + 2 * kh;
                v2f a, bm;
                {
                    int a0 = (c0     < CIN) ? c0     : CIN - 1;
                    int a1 = (c0 + 1 < CIN) ? c0 + 1 : CIN - 1;
                    a.x = wrow[a0] * ((c0     < CIN) ? 1.f : 0.f);
                    a.y = wrow[a1] * ((c0 + 1 < CIN) ? 1.f : 0.f);
                }
                float f[2];
                #pragma unroll
                for (int u = 0; u < 2; ++u) {
                    const int c = c0 + u;
                    const bool valid = c < CIN;
                    const bool isN   = c < CH;
                    int cm = isN ? c : c - CH;
                    cm = (cm < CH) ? cm : CH - 1;         // clamp for invalid lanes
                    const float* pb = isN ? ptn : ptc;    // cndmask'd address
                    float v  = pb[cm];                    // unconditional load
                    float ct = ptc[cm];
                    float fv = v - (isN ? ct : 0.f);
                    f[u] = valid ? fv : 0.f;
                }
                bm.x = f[0]; bm.y = f[1];
                acc = __builtin_amdgcn_wmma_f32_16x16x4_f32(false, a, false, bm,
                                                            (short)0, acc, false, false);
            }
            if (PASS == 0) {
                #pragma unroll
                for (int i = 0; i < 8; ++i) { float v = acc[i]; s1[i] += v; s2[i] += v * v; }
            } else {
                #pragma unroll
                for (int i = 0; i < 8; ++i) {
                    float v = acc[i] * sc[i] + sh[i];
                    v = (v >= 0.f) ? v : 0.2f * v;
                    mx[i] = fmaxf(mx[i], v);
                }
            }
        }
    }

    if (PASS == 0) {
        #pragma unroll
        for (int i = 0; i < 8; ++i) {
            float r1 = s1[i], r2 = s2[i];
            for (int off = 8; off > 0; off >>= 1) {
                r1 += __shfl_xor(r1, off, 32);
                r2 += __shfl_xor(r2, off, 32);
            }
            if ((l & 15) == 0) {
                int r = ro + 8 * kh + i;
                atomicAdd(&stats[2 * r],     r1);
                atomicAdd(&stats[2 * r + 1], r2);
            }
        }
    } else {
        float* po = outT + (size_t)nIdx * COUT + ro + 8 * kh;   // contiguous 8 floats
        #pragma unroll
        for (int i = 0; i < 8; ++i) po[i] = mx[i];
    }
}

// ---------------------------------------------------------------------------
// Final 1024x512 GEMM over concat(x1t..x4t). Branchless pointer select per
// k-step; b64 loads for both operands. PASS 1 writes the (B,1024,N) output.
// ---------------------------------------------------------------------------
template<int PASS>
__global__ __launch_bounds__(128)
void final_gemm_kernel(const float* __restrict__ x1t, const float* __restrict__ x2t,
                       const float* __restrict__ x3t, const float* __restrict__ x4t,
                       const float* __restrict__ w, float* __restrict__ stats,
                       const float* __restrict__ ss, float* __restrict__ out) {
    constexpr int CIN = 512, COUT = 1024;
    constexpr int CT = (BATCH * NPTS) / 16;

    const int l  = threadIdx.x & 31;
    const int W  = blockIdx.x * 4 + (threadIdx.x >> 5);
    const int ro = (W / CT) * 16;
    const int nIdx = (W % CT) * 16 + (l & 15);
    const int b = nIdx >> NBITS, n = nIdx & (NPTS - 1);
    const int kh = l >> 4;
    const float* wrow = w + (size_t)(ro + (l & 15)) * CIN;

    const float* r1 = x1t + (size_t)nIdx * 64;
    const float* r2 = x2t + (size_t)nIdx * 64;
    const float* r3 = x3t + (size_t)nIdx * 128;
    const float* r4 = x4t + (size_t)nIdx * 256;

    v8f acc = {};
    #pragma unroll
    for (int s = 0; s < CIN / 4; ++s) {
        const int c0 = 4 * s + 2 * kh;
        v2f a = *(const v2f*)(wrow + c0);
        const float* p = (c0 < 64)  ? r1 + c0
                       : (c0 < 128) ? r2 + (c0 - 64)
                       : (c0 < 256) ? r3 + (c0 - 128)
                                    : r4 + (c0 - 256);
        v2f bm = *(const v2f*)p;
        acc = __builtin_amdgcn_wmma_f32_16x16x4_f32(false, a, false, bm,
                                                    (short)0, acc, false, false);
    }

    if (PASS == 0) {
        #pragma unroll
        for (int i = 0; i < 8; ++i) {
            float r1s = acc[i], r2s = acc[i] * acc[i];
            for (int off = 8; off > 0; off >>= 1) {
                r1s += __shfl_xor(r1s, off, 32);
                r2s += __shfl_xor(r2s, off, 32);
            }
            if ((l & 15) == 0) {
                int r = ro + 8 * kh + i;
                atomicAdd(&stats[2 * r],     r1s);
                atomicAdd(&stats[2 * r + 1], r2s);
            }
        }
    } else {
        #pragma unroll
        for (int i = 0; i < 8; ++i) {
            int r = ro + 8 * kh + i;
            float v = acc[i] * ss[2 * r] + ss[2 * r + 1];
            v = (v >= 0.f) ? v : 0.2f * v;
            out[(size_t)b * COUT * NPTS + (size_t)r * NPTS + n] = v;
        }
    }
}

// ---------------------------------------------------------------------------
// Helpers: zero a buffer; turn (sum,sumsq,g,b) into (scale,shift).
// ---------------------------------------------------------------------------
__global__ void zero_kernel(float* __restrict__ p, int n) {
    int t = blockIdx.x * blockDim.x + threadIdx.x;
    if (t < n) p[t] = 0.f;
}

__global__ void finalize_kernel(const float* __restrict__ stats,
                                const float* __restrict__ g, const float* __restrict__ bb,
                                float* __restrict__ ss, int C, float invCount) {
    int o = blockIdx.x * blockDim.x + threadIdx.x;
    if (o >= C) return;
    float mean  = stats[2 * o] * invCount;
    float var   = stats[2 * o + 1] * invCount - mean * mean;
    float scale = g[o] * rsqrtf(var + EPSV);
    ss[2 * o]     = scale;
    ss[2 * o + 1] = bb[o] - mean * scale;
}

// ---------------------------------------------------------------------------
extern "C" void kernel_launch(void* const* d_in, const int* in_sizes, int n_in,
                              void* d_out, int out_size, void* d_ws, size_t ws_size,
                              hipStream_t stream) {
    const float* x  = (const float*)d_in[0];
    const float* w1 = (const float*)d_in[1];
    const float* w2 = (const float*)d_in[2];
    const float* w3 = (const float*)d_in[3];
    const float* w4 = (const float*)d_in[4];
    const float* w5 = (const float*)d_in[5];
    const float* g1 = (const float*)d_in[6];  const float* b1 = (const float*)d_in[7];
    const float* g2 = (const float*)d_in[8];  const float* b2 = (const float*)d_in[9];
    const float* g3 = (const float*)d_in[10]; const float* b3 = (const float*)d_in[11];
    const float* g4 = (const float*)d_in[12]; const float* b4 = (const float*)d_in[13];
    const float* g5 = (const float*)d_in[14]; const float* b5 = (const float*)d_in[15];
    float* out = (float*)d_out;

    char* ws = (char*)d_ws;
    size_t off = 0;
    auto alloc = [&](size_t bytes) -> void* {
        void* p = ws + off;
        off += (bytes + 255) & ~(size_t)255;
        return p;
    };
    float* xx    = (float*)alloc((size_t)BATCH * NPTS * 4);
    int*   idxb  = (int*)  alloc((size_t)BATCH * NPTS * KNN * 4);
    float* xt0   = (float*)alloc((size_t)BATCH * NPTS * 3   * 4);
    float* x1t   = (float*)alloc((size_t)BATCH * NPTS * 64  * 4);
    float* x2t   = (float*)alloc((size_t)BATCH * NPTS * 64  * 4);
    float* x3t   = (float*)alloc((size_t)BATCH * NPTS * 128 * 4);
    float* x4t   = (float*)alloc((size_t)BATCH * NPTS * 256 * 4);
    float* stats = (float*)alloc(2 * 1024 * 4);
    float* ssb   = (float*)alloc(2 * 1024 * 4);

    const float invEdge  = 1.0f / (float)((size_t)BATCH * NPTS * KNN);
    const float invFinal = 1.0f / (float)((size_t)BATCH * NPTS);
    const int knnBlocks = (BATCH * NPTS) / 4;                // 4096
    const int ptBlocks  = (BATCH * NPTS + 255) / 256;

    // ---- EdgeConv block 1: x(3) -> x1t(64) ----
    transpose3_kernel<<<ptBlocks, 256, 0, stream>>>(x, xt0);
    xx_kernel<<<ptBlocks, 256, 0, stream>>>(xt0, xx, 3);
    knn_kernel<3><<<knnBlocks, 128, 0, stream>>>(xt0, xx, idxb);
    zero_kernel<<<1, 128, 0, stream>>>(stats, 128);
    edge_conv_kernel<3, 64, 0><<<(64 / 16) * 1024 / 4, 128, 0, stream>>>(xt0, w1, idxb, stats, nullptr, nullptr);
    finalize_kernel<<<1, 64, 0, stream>>>(stats, g1, b1, ssb, 64, invEdge);
    edge_conv_kernel<3, 64, 1><<<(64 / 16) * 1024 / 4, 128, 0, stream>>>(xt0, w1, idxb, nullptr, ssb, x1t);

    // ---- EdgeConv block 2: x1t(64) -> x2t(64) ----
    xx_kernel<<<ptBlocks, 256, 0, stream>>>(x1t, xx, 64);
    knn_kernel<64><<<knnBlocks, 128, 0, stream>>>(x1t, xx, idxb);
    zero_kernel<<<1, 128, 0, stream>>>(stats, 128);
    edge_conv_kernel<64, 64, 0><<<(64 / 16) * 1024 / 4, 128, 0, stream>>>(x1t, w2, idxb, stats, nullptr, nullptr);
    finalize_kernel<<<1, 64, 0, stream>>>(stats, g2, b2, ssb, 64, invEdge);
    edge_conv_kernel<64, 64, 1><<<(64 / 16) * 1024 / 4, 128, 0, stream>>>(x1t, w2, idxb, nullptr, ssb, x2t);

    // ---- EdgeConv block 3: x2t(64) -> x3t(128) ----
    xx_kernel<<<ptBlocks, 256, 0, stream>>>(x2t, xx, 64);
    knn_kernel<64><<<knnBlocks, 128, 0, stream>>>(x2t, xx, idxb);
    zero_kernel<<<1, 256, 0, stream>>>(stats, 256);
    edge_conv_kernel<64, 128, 0><<<(128 / 16) * 1024 / 4, 128, 0, stream>>>(x2t, w3, idxb, stats, nullptr, nullptr);
    finalize_kernel<<<1, 128, 0, stream>>>(stats, g3, b3, ssb, 128, invEdge);
    edge_conv_kernel<64, 128, 1><<<(128 / 16) * 1024 / 4, 128, 0, stream>>>(x2t, w3, idxb, nullptr, ssb, x3t);

    // ---- EdgeConv block 4: x3t(128) -> x4t(256) ----
    xx_kernel<<<ptBlocks, 256, 0, stream>>>(x3t, xx, 128);
    knn_kernel<128><<<knnBlocks, 128, 0, stream>>>(x3t, xx, idxb);
    zero_kernel<<<2, 256, 0, stream>>>(stats, 512);
    edge_conv_kernel<128, 256, 0><<<(256 / 16) * 1024 / 4, 128, 0, stream>>>(x3t, w4, idxb, stats, nullptr, nullptr);
    finalize_kernel<<<1, 256, 0, stream>>>(stats, g4, b4, ssb, 256, invEdge);
    edge_conv_kernel<128, 256, 1><<<(256 / 16) * 1024 / 4, 128, 0, stream>>>(x3t, w4, idxb, nullptr, ssb, x4t);

    // ---- Final: concat(512) -> 1024, BN + lrelu, (B,1024,N) output ----
    zero_kernel<<<8, 256, 0, stream>>>(stats, 2048);
    final_gemm_kernel<0><<<(1024 / 16) * 1024 / 4, 128, 0, stream>>>(x1t, x2t, x3t, x4t, w5, stats, nullptr, nullptr);
    finalize_kernel<<<4, 256, 0, stream>>>(stats, g5, b5, ssb, 1024, invFinal);
    final_gemm_kernel<1><<<(1024 / 16) * 1024 / 4, 128, 0, stream>>>(x1t, x2t, x3t, x4t, w5, nullptr, ssb, out);
}